// QuantumLayer_9835475108036
// MI455X (gfx1250) — compile-verified
//
#include <hip/hip_runtime.h>

// out[b][0] = prod_{j=1..7} cos(pi*clip(x[b][j],0,1))
// out[b][i] = prod_{j=0..i} cos(pi*clip(x[b][j],0,1))   for i = 1..7
//
// Derivation: RY embedding makes qubit bits independent with
// E[(-1)^b_j] = cos(theta_j); the CNOT chain maps wire-0's bit to
// b1^...^b7 and wire-i's bit to b0^...^bi; independence factorizes the
// expectation into products of cos(theta_j).

__global__ __launch_bounds__(256) void quantum_chain_z_kernel(
    const float* __restrict__ x, float* __restrict__ out, int B) {
  int i = blockIdx.x * blockDim.x + threadIdx.x;
  if (i >= B) return;

  // Two coalesced 16B vector loads per sample -> global_load_b128 x2.
  const float4* __restrict__ xin = reinterpret_cast<const float4*>(x) + (size_t)i * 2;
  float4 a = xin[0];
  float4 b = xin[1];

  float v[8] = {a.x, a.y, a.z, a.w, b.x, b.y, b.z, b.w};
  float z[8];
#pragma unroll
  for (int j = 0; j < 8; ++j) {
    float t = __builtin_fminf(__builtin_fmaxf(v[j], 0.0f), 1.0f);
    z[j] = cospif(t);  // cos(pi * t), exact pi scaling, ~1 ulp
  }

  float o[8];
  // Prefix products for wires 1..7.
  float p = z[0];
#pragma unroll
  for (int j = 1; j < 8; ++j) {
    p *= z[j];
    o[j] = p;
  }
  // Wire 0: product of z1..z7 (no division so z0 == 0 is safe).
  float q = z[7];
#pragma unroll
  for (int j = 6; j >= 1; --j) q *= z[j];
  o[0] = q;

  // Two coalesced 16B vector stores -> global_store_b128 x2.
  float4* __restrict__ op = reinterpret_cast<float4*>(out) + (size_t)i * 2;
  op[0] = make_float4(o[0], o[1], o[2], o[3]);
  op[1] = make_float4(o[4], o[5], o[6], o[7]);
}

extern "C" void kernel_launch(void* const* d_in, const int* in_sizes, int n_in,
                              void* d_out, int out_size, void* d_ws, size_t ws_size,
                              hipStream_t stream) {
  (void)n_in; (void)out_size; (void)d_ws; (void)ws_size;
  const float* x = reinterpret_cast<const float*>(d_in[0]);
  float* out = reinterpret_cast<float*>(d_out);
  int B = in_sizes[0] / 8;  // 262144 samples, 8 qubits each

  const int block = 256;                 // 8 wave32s per block
  const int grid = (B + block - 1) / block;
  quantum_chain_z_kernel<<<grid, block, 0, stream>>>(x, out, B);
}